// TempGT_74311524155546
// MI455X (gfx1250) — compile-verified
//
#include <hip/hip_runtime.h>
#include <hip/hip_bf16.h>
#include <math.h>

// ---------------------------------------------------------------------------
// Model constants (match reference)
// ---------------------------------------------------------------------------
#define Bsz   1024
#define Kn    20
#define Gn    2000
#define Dd    128
#define Tt    100
#define NBf   64
#define THIST 32
#define HEADS 8
#define DH    16
#define LKk   256
#define LN_EPS 1e-5f
#define TWO_PI_64 0.09817477042468103f   // 2*pi/64

typedef float v2f __attribute__((ext_vector_type(2)));
typedef float v8f __attribute__((ext_vector_type(8)));

// ---------------------------------------------------------------------------
// Generic f32 WMMA GEMM:  C[z] = alpha*(A[z] @ B[z]) [+ C] [+ bias] [gelu]
//   A: [M,K] row-major, row stride lda, per-z offset sAz (elements)
//   B: element (k,n) at B + k*bsK + n*bsN   (covers W[N,K] and kproj layouts)
//   C: [M,N], row stride ldc, per-z offset sCz
// One wave (32 threads) computes one 16x16 tile via v_wmma_f32_16x16x4_f32.
// ---------------------------------------------------------------------------
__global__ __launch_bounds__(32) void wmma_gemm(
    const float* __restrict__ A, int lda, long sAz,
    const float* __restrict__ B, int bsK, int bsN, long sBz,
    float* __restrict__ C, int ldc, long sCz,
    int M, int N, int K, float alpha,
    const float* __restrict__ bias, int beta, int gelu)
{
    int tilesN = N >> 4;
    int tn = blockIdx.x % tilesN;
    int tm = blockIdx.x / tilesN;
    int z  = blockIdx.y;
    A += (long)z * sAz;  B += (long)z * sBz;  C += (long)z * sCz;

    int lane = threadIdx.x;      // 0..31
    int half = lane >> 4;        // 0: K={0,1}, 1: K={2,3}
    int l16  = lane & 15;
    int m0 = tm << 4, n0 = tn << 4;

    // A fragment addressing: lane l -> row m0+l16, cols k+2*half, k+2*half+1
    const float* arow = A + (long)(m0 + l16) * lda + 2 * half;
    // B fragment addressing: lane l -> col n0+l16, rows k+2*half, k+2*half+1
    const float* bcol = B + (long)(n0 + l16) * bsN + (long)(2 * half) * bsK;

    v8f acc = {};
    for (int k = 0; k < K; k += 4) {
        v2f a, b;
        a.x = arow[k];
        a.y = arow[k + 1];
        b.x = bcol[(long)k * bsK];
        b.y = bcol[(long)k * bsK + bsK];
        acc = __builtin_amdgcn_wmma_f32_16x16x4_f32(
                  false, a, false, b, (short)0, acc, false, false);
    }

    float bn = bias ? bias[n0 + l16] : 0.0f;
    for (int r = 0; r < 8; ++r) {
        int m = m0 + r + half * 8;
        long idx = (long)m * ldc + n0 + l16;
        float v = alpha * acc[r] + bn;
        if (beta) v += C[idx];
        if (gelu) v = 0.5f * v * (1.0f + erff(v * 0.70710678118654752f));
        C[idx] = v;
    }
}

// ---------------------------------------------------------------------------
// prep: G[j], h[t] for collapsed FFT chain; fused bias for edge MLP chain
// ---------------------------------------------------------------------------
__global__ __launch_bounds__(128) void prep_kernel(
    const float* __restrict__ agg,   // fft_agg_w [64]
    const int*   __restrict__ bidx_p,
    const float* __restrict__ efw,   // edge_final_w [20]
    const float* __restrict__ emb_b, // edge_mlp_b [128]
    const float* __restrict__ efb,   // edge_final_b [1]
    float* __restrict__ hout,        // [32]
    float* __restrict__ bias1)       // [128]
{
    __shared__ float sG[64];
    int tid = threadIdx.x;
    int bidx = *bidx_p; if (bidx > 64) bidx = 64; if (bidx < 0) bidx = 0;
    if (tid < 64) {
        float g = 0.0f;
        if (tid < bidx) {
            for (int t = 0; t < bidx; ++t) {
                int r = (t * tid) & 63;
                g += agg[t] * cosf(TWO_PI_64 * (float)r);
            }
            g *= (1.0f / 64.0f);
        }
        sG[tid] = g;
    }
    __syncthreads();
    if (tid < 32) {
        float h = 0.0f;
        if (tid < bidx) {
            for (int j = 0; j < 64; ++j) {
                int r = (j * tid) & 63;
                h += sG[j] * cosf(TWO_PI_64 * (float)r);
            }
        }
        hout[tid] = h;
    }
    // bias1[e] = (sum_k efw[k]) * edge_mlp_b[e] + edge_final_b
    float sw = 0.0f;
    for (int k = 0; k < Kn; ++k) sw += efw[k];
    bias1[tid] = sw * emb_b[tid] + efb[0];
}

// ---------------------------------------------------------------------------
// edge branch pre-reduction over K: rin[b,c] = sum_k efw[k]*in[b,k,c]
//   c<128: edge features;  c in [128,228): masked time encoding
// ---------------------------------------------------------------------------
__global__ __launch_bounds__(256) void edge_reduce_kernel(
    const float* __restrict__ nit, const float* __restrict__ ntimes,
    const int* __restrict__ nnids, const int* __restrict__ neids,
    const float* __restrict__ eraw, const float* __restrict__ efw,
    const float* __restrict__ tfreq, const float* __restrict__ tbias,
    float* __restrict__ rin)
{
    int b = blockIdx.x;
    int c = threadIdx.x;
    if (c >= Dd + Tt) return;
    float t0 = nit[b];
    float acc = 0.0f;
    for (int k = 0; k < Kn; ++k) {
        float w = efw[k];
        if (c < Dd) {
            int eid = neids[b * Kn + k];
            acc += w * eraw[(long)eid * Dd + c];
        } else {
            int nid = nnids[b * Kn + k];
            float dt = t0 - ntimes[b * Kn + k];
            int tc = c - Dd;
            float tf = (nid == 0) ? 0.0f : cosf(dt * tfreq[tc] + tbias[tc]);
            acc += w * tf;
        }
    }
    rin[(long)b * (Dd + Tt) + c] = acc;
}

// ---------------------------------------------------------------------------
// time-gap branch: softmax over validity mask == (1/nvalid for valid),
// out_nf[b] = (1/G) * sum_g score*feat + node_raw[node_ids[b]]
// 4 waves/block; each wave loads one full 128-f row per iteration (float4/lane)
// ---------------------------------------------------------------------------
__global__ __launch_bounds__(128) void timegap_kernel(
    const int* __restrict__ tgids, const float* __restrict__ nraw,
    const int* __restrict__ node_ids, float* __restrict__ out_nf)
{
    __shared__ int s_cnt;
    __shared__ float s_part[4][Dd];
    int b = blockIdx.x;
    const int* ids = tgids + (long)b * Gn;
    int tid = threadIdx.x, lane = tid & 31, w = tid >> 5;
    if (tid == 0) s_cnt = 0;
    __syncthreads();
    int cnt = 0;
    for (int g = tid; g < Gn; g += 128) cnt += (ids[g] > 0) ? 1 : 0;
    for (int off = 16; off; off >>= 1) cnt += __shfl_xor(cnt, off, 32);
    if (lane == 0) atomicAdd(&s_cnt, cnt);
    __syncthreads();
    int nv = s_cnt;
    float wv, wi;
    if (nv > 0) { wv = 1.0f / (float)nv; wi = 0.0f; }
    else        { wv = wi = 1.0f / (float)Gn; }
    float4 acc = make_float4(0.f, 0.f, 0.f, 0.f);
    for (int g = w; g < Gn; g += 4) {
        int id = ids[g];
        float wt = (id > 0) ? wv : wi;
        if (wt != 0.0f) {
            const float4* row = (const float4*)(nraw + (long)id * Dd);
            float4 v = row[lane];
            acc.x += wt * v.x; acc.y += wt * v.y;
            acc.z += wt * v.z; acc.w += wt * v.w;
        }
    }
    ((float4*)s_part[w])[lane] = acc;
    __syncthreads();
    int d = tid;
    float v = s_part[0][d] + s_part[1][d] + s_part[2][d] + s_part[3][d];
    v = v * (1.0f / (float)Gn) + nraw[(long)node_ids[b] * Dd + d];
    out_nf[(long)b * Dd + d] = v;
}

// ---------------------------------------------------------------------------
// collapsed-FFT per-node reductions: p = sum_t a[t]*x, q = sum_t h[t]*x
// ---------------------------------------------------------------------------
__global__ __launch_bounds__(128) void pe_reduce_kernel(
    const float* __restrict__ pe_table, const int* __restrict__ node_ids,
    const float* __restrict__ agg, const float* __restrict__ hvec,
    float* __restrict__ pbuf, float* __restrict__ qbuf)
{
    int b = blockIdx.x, d = threadIdx.x;
    const float* x = pe_table + (long)node_ids[b] * THIST * Dd + d;
    float p = 0.0f, q = 0.0f;
    for (int t = 0; t < THIST; ++t) {
        float xv = x[t * Dd];
        p += agg[t] * xv;
        q += hvec[t] * xv;
    }
    pbuf[(long)b * Dd + d] = p;
    qbuf[(long)b * Dd + d] = q;
}

// x0[b,d] = p[b,d] + node_final_b[d]
__global__ __launch_bounds__(128) void x0_init_kernel(
    const float* __restrict__ p, const float* __restrict__ nfb,
    float* __restrict__ x)
{
    long i = (long)blockIdx.x * Dd + threadIdx.x;
    x[i] = p[i] + nfb[threadIdx.x];
}

// ---------------------------------------------------------------------------
// LayerNorm over last dim (128); one wave per row, float4 per lane
// ---------------------------------------------------------------------------
__global__ __launch_bounds__(128) void layernorm_kernel(
    const float* __restrict__ x, const float* __restrict__ g,
    const float* __restrict__ bta, float* __restrict__ out)
{
    int lane = threadIdx.x & 31, w = threadIdx.x >> 5;
    int row = blockIdx.x * 4 + w;
    const float4* xr = (const float4*)(x + (long)row * Dd);
    float4 v = xr[lane];
    float s = v.x + v.y + v.z + v.w;
    for (int off = 16; off; off >>= 1) s += __shfl_xor(s, off, 32);
    float m = s * (1.0f / (float)Dd);
    float dx = v.x - m, dy = v.y - m, dz = v.z - m, dw = v.w - m;
    float vs = dx * dx + dy * dy + dz * dz + dw * dw;
    for (int off = 16; off; off >>= 1) vs += __shfl_xor(vs, off, 32);
    float inv = rsqrtf(vs * (1.0f / (float)Dd) + LN_EPS);
    float4 gv = ((const float4*)g)[lane];
    float4 bv = ((const float4*)bta)[lane];
    float4 o;
    o.x = dx * inv * gv.x + bv.x;
    o.y = dy * inv * gv.y + bv.y;
    o.z = dz * inv * gv.z + bv.z;
    o.w = dw * inv * gv.w + bv.w;
    ((float4*)(out + (long)row * Dd))[lane] = o;
}

// kproj[k,d] = sum_n kv[n,d] * proj_k[n,k]   (k<256, d<16, n<1024)
__global__ __launch_bounds__(128) void kproj_kernel(
    const float* __restrict__ kv, const float* __restrict__ pk,
    float* __restrict__ kproj)
{
    __shared__ float s[8][DH];
    int k = blockIdx.x;
    int d = threadIdx.x & 15, seg = threadIdx.x >> 4;
    float acc = 0.0f;
    for (int i = 0; i < 128; ++i) {
        int n = seg * 128 + i;
        acc += kv[n * DH + d] * pk[(long)n * LKk + k];
    }
    s[seg][d] = acc;
    __syncthreads();
    if (threadIdx.x < DH) {
        float v = 0.0f;
        for (int sg = 0; sg < 8; ++sg) v += s[sg][threadIdx.x];
        kproj[k * DH + threadIdx.x] = v;
    }
}

// softmax over rows of 256; one wave per row (8 elems/lane)
__global__ __launch_bounds__(128) void softmax256_kernel(float* __restrict__ x)
{
    int lane = threadIdx.x & 31, w = threadIdx.x >> 5;
    int row = blockIdx.x * 4 + w;
    float4* p = (float4*)(x + (long)row * LKk);
    float4 a = p[lane], b = p[lane + 32];
    float mx = fmaxf(fmaxf(fmaxf(a.x, a.y), fmaxf(a.z, a.w)),
                     fmaxf(fmaxf(b.x, b.y), fmaxf(b.z, b.w)));
    for (int off = 16; off; off >>= 1) mx = fmaxf(mx, __shfl_xor(mx, off, 32));
    a.x = expf(a.x - mx); a.y = expf(a.y - mx);
    a.z = expf(a.z - mx); a.w = expf(a.w - mx);
    b.x = expf(b.x - mx); b.y = expf(b.y - mx);
    b.z = expf(b.z - mx); b.w = expf(b.w - mx);
    float s = a.x + a.y + a.z + a.w + b.x + b.y + b.z + b.w;
    for (int off = 16; off; off >>= 1) s += __shfl_xor(s, off, 32);
    float inv = 1.0f / s;
    a.x *= inv; a.y *= inv; a.z *= inv; a.w *= inv;
    b.x *= inv; b.y *= inv; b.z *= inv; b.w *= inv;
    p[lane] = a; p[lane + 32] = b;
}

// ---------------------------------------------------------------------------
// host-side launcher
// ---------------------------------------------------------------------------
static void launch_gemm(hipStream_t s,
                        const float* A, int lda, long sAz,
                        const float* B, int bsK, int bsN, long sBz,
                        float* C, int ldc, long sCz,
                        int M, int N, int K, float alpha,
                        const float* bias, int beta, int gelu, int Z)
{
    dim3 grid((M / 16) * (N / 16), Z, 1);
    wmma_gemm<<<grid, 32, 0, s>>>(A, lda, sAz, B, bsK, bsN, sBz,
                                  C, ldc, sCz, M, N, K, alpha, bias, beta, gelu);
}

extern "C" void kernel_launch(void* const* d_in, const int* in_sizes, int n_in,
                              void* d_out, int out_size, void* d_ws, size_t ws_size,
                              hipStream_t stream) {
    // inputs (setup_inputs order)
    const float* nit   = (const float*)d_in[0];
    const float* ntime = (const float*)d_in[1];
    const float* petab = (const float*)d_in[2];
    const float* nraw  = (const float*)d_in[3];
    const float* eraw  = (const float*)d_in[4];
    const float* fftw  = (const float*)d_in[5];
    const float* fagg  = (const float*)d_in[6];
    const float* tfreq = (const float*)d_in[7];
    const float* tbias = (const float*)d_in[8];
    const float* emw   = (const float*)d_in[9];
    const float* emb   = (const float*)d_in[10];
    const float* efw   = (const float*)d_in[11];
    const float* efb   = (const float*)d_in[12];
    const float* nfw   = (const float*)d_in[13];
    const float* nfb   = (const float*)d_in[14];
    const float* ln1g  = (const float*)d_in[15];
    const float* ln1b  = (const float*)d_in[16];
    const float* wq    = (const float*)d_in[17];
    const float* wk    = (const float*)d_in[18];
    const float* pk    = (const float*)d_in[19];
    const float* wo    = (const float*)d_in[20];
    const float* bo    = (const float*)d_in[21];
    const float* ln2g  = (const float*)d_in[22];
    const float* ln2b  = (const float*)d_in[23];
    const float* ffw1  = (const float*)d_in[24];
    const float* ffb1  = (const float*)d_in[25];
    const float* ffw2  = (const float*)d_in[26];
    const float* ffb2  = (const float*)d_in[27];
    const int* node_ids = (const int*)d_in[28];
    const int* nnids    = (const int*)d_in[29];
    const int* neids    = (const int*)d_in[30];
    const int* tgids    = (const int*)d_in[31];
    const int* bidx     = (const int*)d_in[32];

    // workspace layout (floats)
    float* ws    = (float*)d_ws;
    float* rin   = ws;                 // 1024*228
    float* comb  = rin   + Bsz * (Dd + Tt);
    float* outnf = comb  + Bsz * Dd;
    float* pbuf  = outnf + Bsz * Dd;
    float* qpe   = pbuf  + Bsz * Dd;
    float* hvec  = qpe   + Bsz * Dd;   // 32 (padded to 64)
    float* bias1 = hvec  + 64;         // 128
    float* xbuf  = bias1 + 128;
    float* h1    = xbuf  + Bsz * Dd;
    float* qb    = h1    + Bsz * Dd;
    float* kvb   = qb    + Bsz * Dd;
    float* kpj   = kvb   + Bsz * DH;
    float* dots  = kpj   + LKk * DH;
    float* obuf  = dots  + (long)HEADS * Bsz * LKk;
    float* h2    = obuf  + Bsz * Dd;
    float* ffb   = h2    + Bsz * Dd;   // 1024*512

    // ---- phase 1: branches ----
    prep_kernel<<<1, 128, 0, stream>>>(fagg, bidx, efw, emb, efb, hvec, bias1);
    edge_reduce_kernel<<<Bsz, 256, 0, stream>>>(nit, ntime, nnids, neids, eraw,
                                                efw, tfreq, tbias, rin);
    timegap_kernel<<<Bsz, 128, 0, stream>>>(tgids, nraw, node_ids, outnf);
    pe_reduce_kernel<<<Bsz, 128, 0, stream>>>(petab, node_ids, fagg, hvec, pbuf, qpe);

    // comb = rin @ edge_mlp_w^T + (sw*edge_mlp_b + edge_final_b)
    launch_gemm(stream, rin, Dd + Tt, 0, emw, 1, Dd + Tt, 0,
                comb, Dd, 0, Bsz, Dd, Dd + Tt, 1.0f, bias1, 0, 0, 1);

    // x0 = p + node_final_b + comb@Wn1^T + out_nf@Wn2^T + q@Wfilter^T
    x0_init_kernel<<<Bsz, 128, 0, stream>>>(pbuf, nfb, xbuf);
    launch_gemm(stream, comb,  Dd, 0, nfw,       1, 2 * Dd, 0,
                xbuf, Dd, 0, Bsz, Dd, Dd, 1.0f, nullptr, 1, 0, 1);
    launch_gemm(stream, outnf, Dd, 0, nfw + Dd,  1, 2 * Dd, 0,
                xbuf, Dd, 0, Bsz, Dd, Dd, 1.0f, nullptr, 1, 0, 1);
    launch_gemm(stream, qpe,   Dd, 0, fftw,      1, Dd,     0,
                xbuf, Dd, 0, Bsz, Dd, Dd, 1.0f, nullptr, 1, 0, 1);

    // ---- phase 2: Linformer layers ----
    const float scale = 0.25f;  // DH^-0.5
    for (int l = 0; l < 2; ++l) {
        layernorm_kernel<<<Bsz / 4, 128, 0, stream>>>(xbuf, ln1g + l * Dd,
                                                      ln1b + l * Dd, h1);
        // q = h1 @ wq^T
        launch_gemm(stream, h1, Dd, 0, wq + (long)l * Dd * Dd, 1, Dd, 0,
                    qb, Dd, 0, Bsz, Dd, Dd, 1.0f, nullptr, 0, 0, 1);
        // kv = h1 @ wk^T  [1024,16]
        launch_gemm(stream, h1, Dd, 0, wk + (long)l * DH * Dd, 1, Dd, 0,
                    kvb, DH, 0, Bsz, DH, Dd, 1.0f, nullptr, 0, 0, 1);
        // kproj = proj_k^T @ kv  [256,16]
        kproj_kernel<<<LKk, 128, 0, stream>>>(kvb, pk + (long)l * Bsz * LKk, kpj);
        // dots[h] = (q_h @ kproj^T) * scale   [1024,256] x 8 heads
        launch_gemm(stream, qb, Dd, DH, kpj, 1, DH, 0,
                    dots, LKk, (long)Bsz * LKk,
                    Bsz, LKk, DH, scale, nullptr, 0, 0, HEADS);
        softmax256_kernel<<<HEADS * Bsz / 4, 128, 0, stream>>>(dots);
        // o[h] = attn_h @ kproj  -> obuf[n, h*16+d]
        launch_gemm(stream, dots, LKk, (long)Bsz * LKk, kpj, DH, 1, 0,
                    obuf, Dd, DH, Bsz, DH, LKk, 1.0f, nullptr, 0, 0, HEADS);
        // x += o @ wo^T + bo
        launch_gemm(stream, obuf, Dd, 0, wo + (long)l * Dd * Dd, 1, Dd, 0,
                    xbuf, Dd, 0, Bsz, Dd, Dd, 1.0f, bo + l * Dd, 1, 0, 1);
        // FFN
        layernorm_kernel<<<Bsz / 4, 128, 0, stream>>>(xbuf, ln2g + l * Dd,
                                                      ln2b + l * Dd, h2);
        launch_gemm(stream, h2, Dd, 0, ffw1 + (long)l * 4 * Dd * Dd, 1, Dd, 0,
                    ffb, 4 * Dd, 0, Bsz, 4 * Dd, Dd, 1.0f,
                    ffb1 + l * 4 * Dd, 0, 1, 1);
        launch_gemm(stream, ffb, 4 * Dd, 0, ffw2 + (long)l * 4 * Dd * Dd, 1, 4 * Dd, 0,
                    xbuf, Dd, 0, Bsz, Dd, 4 * Dd, 1.0f,
                    ffb2 + l * Dd, 1, 0, 1);
    }

    hipMemcpyAsync(d_out, xbuf, (size_t)Bsz * Dd * sizeof(float),
                   hipMemcpyDeviceToDevice, stream);
}